// Conductor_18966575579282
// MI455X (gfx1250) — compile-verified
//
#include <hip/hip_runtime.h>

// ---------------------------------------------------------------------------
// Sizes from the reference
// ---------------------------------------------------------------------------
#define CC     512      // C
#define TT     128      // T frames
#define VV     8        // voices
#define LL     3        // MLP depth
#define KK     16       // scan steps
#define NOSC   64       // oscillators
#define SPF    256
#define NC     257      // SPF+1 rfft bins of a 512 frame
#define FRLEN  4096     // K*SPF
#define TOTAL  32768    // T*SPF

#define ACT_NONE  0
#define ACT_LRELU 1
#define ACT_SIN30 2

typedef float v2f __attribute__((ext_vector_type(2)));
typedef float v8f __attribute__((ext_vector_type(8)));

__device__ __forceinline__ int clampi(int v, int hi) { return v < 0 ? 0 : (v > hi ? hi : v); }

// ---------------------------------------------------------------------------
// Generic f32 WMMA GEMM:  Out[b] = act( A[b] @ W[sel(b)] + bias[sel(b)] )
// One wave per 16x16 output tile, V_WMMA_F32_16X16X4_F32 in the K loop.
// Fast path for fully in-bounds tiles: unguarded pointer-bumped loads.
// Ragged tiles / K tails use clamped-address loads + select-zero, which is
// branch-free (v_cndmask) and keeps EXEC all-ones throughout (WMMA rule).
// accMode: 0 = store(bias+act), 1 = Out += tile, 2 = Out -= tile
// ---------------------------------------------------------------------------
__global__ __launch_bounds__(32)
void wmma_gemm_f32(const float* __restrict__ A, int lda, long aBatch,
                   const float* __restrict__ W, int ldb, long wBatch,
                   const int* __restrict__ vsel,
                   const float* __restrict__ bias, long biasBatch,
                   float* __restrict__ Out, int ldo, long oBatch,
                   int M, int N, int K, int act, int accMode)
{
    int tn = blockIdx.x, tm = blockIdx.y, b = blockIdx.z;
    int lane = threadIdx.x;
    int hi = lane >> 4;          // lane half selects K pair
    int l  = lane & 15;
    int sel = vsel ? vsel[b] : 0;
    const float* Ab = A + (long)b * aBatch;
    const float* Wb = W + (long)sel * wBatch;
    const float* Bb = bias ? (bias + (long)sel * biasBatch) : nullptr;
    float* Ob = Out + (long)b * oBatch;

    int m0 = tm * 16, n0 = tn * 16;
    int row = m0 + l;
    int col = n0 + l;
    int n = n0 + l;
    v8f acc = {};
    bool full = (m0 + 16 <= M) && (n0 + 16 <= N);
    int kmain = K & ~3;

    if (full) {
        // ---- fast path: no guards at all in the main K body ----
        const float* pa  = Ab + (long)row * lda + 2 * hi;
        const float* pb0 = Wb + (long)(2 * hi) * ldb + col;
        const float* pb1 = pb0 + ldb;
        long pbStep = 4L * ldb;
        for (int k0 = 0; k0 < kmain; k0 += 4) {
            v2f av, bv;
            av.x = pa[0];
            av.y = pa[1];
            bv.x = pb0[0];
            bv.y = pb1[0];
            acc = __builtin_amdgcn_wmma_f32_16x16x4_f32(false, av, false, bv,
                                                        (short)0, acc, false, false);
            pa  += 4;
            pb0 += pbStep;
            pb1 += pbStep;
        }
        if (kmain < K) {                     // K tail (e.g. K = 257), branch-free
            int ka = kmain + 2 * hi;
            int k0c = clampi(ka, K - 1), k1c = clampi(ka + 1, K - 1);
            float a0 = Ab[(long)row * lda + k0c];
            float a1 = Ab[(long)row * lda + k1c];
            float b0 = Wb[(long)k0c * ldb + col];
            float b1 = Wb[(long)k1c * ldb + col];
            v2f av, bv;
            av.x = (ka     < K) ? a0 : 0.f;
            av.y = (ka + 1 < K) ? a1 : 0.f;
            bv.x = (ka     < K) ? b0 : 0.f;
            bv.y = (ka + 1 < K) ? b1 : 0.f;
            acc = __builtin_amdgcn_wmma_f32_16x16x4_f32(false, av, false, bv,
                                                        (short)0, acc, false, false);
        }
        // ---- unguarded store ----
        if (accMode == 0) {
            float bb = Bb ? Bb[n] : 0.f;
#pragma unroll
            for (int r = 0; r < 8; ++r) {
                int m = m0 + r + 8 * hi;
                float v = acc[r] + bb;
                if (act == ACT_LRELU)      v = (v >= 0.f) ? v : 0.2f * v;
                else if (act == ACT_SIN30) v = sinf(30.f * v);
                Ob[(long)m * ldo + n] = v;
            }
        } else if (accMode == 1) {
#pragma unroll
            for (int r = 0; r < 8; ++r) {
                int m = m0 + r + 8 * hi;
                Ob[(long)m * ldo + n] += acc[r];
            }
        } else {
#pragma unroll
            for (int r = 0; r < 8; ++r) {
                int m = m0 + r + 8 * hi;
                Ob[(long)m * ldo + n] -= acc[r];
            }
        }
    } else {
        // ---- ragged path: clamped-address loads, select-zero (branch-free) ----
        int rowC = clampi(row, M - 1);
        int colC = clampi(col, N - 1);
        bool rowOk = row < M, colOk = col < N;
        for (int k0 = 0; k0 < K; k0 += 4) {
            int ka = k0 + 2 * hi;
            int k0c = clampi(ka, K - 1), k1c = clampi(ka + 1, K - 1);
            bool k0ok = ka < K, k1ok = (ka + 1) < K;
            float a0 = Ab[(long)rowC * lda + k0c];
            float a1 = Ab[(long)rowC * lda + k1c];
            float b0 = Wb[(long)k0c * ldb + colC];
            float b1 = Wb[(long)k1c * ldb + colC];
            v2f av, bv;
            av.x = (rowOk && k0ok) ? a0 : 0.f;
            av.y = (rowOk && k1ok) ? a1 : 0.f;
            bv.x = (colOk && k0ok) ? b0 : 0.f;
            bv.y = (colOk && k1ok) ? b1 : 0.f;
            acc = __builtin_amdgcn_wmma_f32_16x16x4_f32(false, av, false, bv,
                                                        (short)0, acc, false, false);
        }
        if (n < N) {
#pragma unroll
            for (int r = 0; r < 8; ++r) {
                int m = m0 + r + 8 * hi;
                if (m < M) {
                    float v = acc[r];
                    if (accMode == 0) {
                        if (Bb) v += Bb[n];
                        if (act == ACT_LRELU)      v = (v >= 0.f) ? v : 0.2f * v;
                        else if (act == ACT_SIN30) v = sinf(30.f * v);
                        Ob[(long)m * ldo + n] = v;
                    } else if (accMode == 1) {
                        Ob[(long)m * ldo + n] += v;
                    } else {
                        Ob[(long)m * ldo + n] -= v;
                    }
                }
            }
        }
    }
}

// ---------------------------------------------------------------------------
// z = mean_t x[0,t,:]
// ---------------------------------------------------------------------------
__global__ void zmean_kernel(const float* __restrict__ x, float* __restrict__ z)
{
    int c = blockIdx.x * blockDim.x + threadIdx.x;
    if (c < CC) {
        float s = 0.f;
        for (int t = 0; t < TT; ++t) s += x[t * CC + c];
        z[c] = s * (1.f / TT);
    }
}

// rooms = softmax(rmlog), mix = sigmoid(mxlog)
__global__ void heads_kernel(const float* __restrict__ rmlog,
                             const float* __restrict__ mxlog,
                             float* __restrict__ rooms, float* __restrict__ mixv)
{
    if (threadIdx.x == 0) {
        float mx = rmlog[0];
        for (int i = 1; i < 8; ++i) mx = fmaxf(mx, rmlog[i]);
        float e[8], s = 0.f;
        for (int i = 0; i < 8; ++i) { e[i] = expf(rmlog[i] - mx); s += e[i]; }
        for (int i = 0; i < 8; ++i) rooms[i] = e[i] / s;
        mixv[0] = 1.f / (1.f + expf(-mxlog[0]));
    }
}

// per-frame routing: softmax over 9, argmax (first max), r, is_sil, vv
__global__ void route_kernel(const float* __restrict__ rlog,
                             int* __restrict__ vv, int* __restrict__ sil,
                             float* __restrict__ rr)
{
    int t = threadIdx.x;
    if (t < TT) {
        const float* row = rlog + t * 9;
        float mx = row[0];
        for (int f = 1; f < 9; ++f) mx = fmaxf(mx, row[f]);
        float e[9], s = 0.f;
        for (int f = 0; f < 9; ++f) { e[f] = expf(row[f] - mx); s += e[f]; }
        int best = 0; float bv = e[0];
        for (int f = 1; f < 9; ++f) if (e[f] > bv) { bv = e[f]; best = f; }
        rr[t]  = bv / s;
        sil[t] = (best == VV) ? 1 : 0;
        vv[t]  = (best == VV) ? 0 : best;
    }
}

// ---------------------------------------------------------------------------
// Fused 16-step recurrence: one workgroup per frame, hh kept in LDS,
// voice-gathered weights streamed from L2 (coalesced along the output dim).
// ---------------------------------------------------------------------------
__global__ __launch_bounds__(256)
void scan_kernel(const float* __restrict__ h, const int* __restrict__ vv,
                 const float* __restrict__ nW, const float* __restrict__ nB,
                 const float* __restrict__ nWo, const float* __restrict__ nBo,
                 const float* __restrict__ gW, const float* __restrict__ gB,
                 const float* __restrict__ gWo, const float* __restrict__ gBo,
                 float* __restrict__ res, float* __restrict__ validOut)
{
    __shared__ float cur[CC], bufA[CC], bufB[CC];
    __shared__ float red[256];
    __shared__ float sc[2];
    __shared__ int aliveS;

    int t = blockIdx.x, tid = threadIdx.x;
    int v = vv[t];
    const float* nWv  = nW  + (long)v * LL * CC * CC;
    const float* nBv  = nB  + (long)v * LL * CC;
    const float* nWov = nWo + (long)v * CC * CC;
    const float* nBov = nBo + (long)v * CC;
    const float* gWv  = gW  + (long)v * LL * CC * CC;
    const float* gBv  = gB  + (long)v * LL * CC;
    const float* gWov = gWo + (long)v * CC * 2;
    const float* gBov = gBo + (long)v * 2;

    float h0 = h[t * CC + tid], h1 = h[t * CC + tid + 256];
    red[tid] = h0 * h0 + h1 * h1;
    __syncthreads();
    for (int off = 128; off > 0; off >>= 1) {
        if (tid < off) red[tid] += red[tid + off];
        __syncthreads();
    }
    float inv = 1.f / (sqrtf(red[0]) + 1e-8f);
    __syncthreads();
    cur[tid] = h0 * inv; cur[tid + 256] = h1 * inv;
    if (tid == 0) aliveS = 1;
    __syncthreads();

    int maxit = TT - t;
    for (int j = 0; j < KK; ++j) {
        // --- n-MLP: 3 hidden (sin30) + linear out ---
        const float* inb = cur; float* outb = bufA;
        for (int layer = 0; layer < LL; ++layer) {
            const float* Wl = nWv + (long)layer * CC * CC;
            const float* Bl = nBv + layer * CC;
            for (int half = 0; half < 2; ++half) {
                int d = tid + half * 256;
                float a = Bl[d];
                for (int c = 0; c < CC; ++c) a += inb[c] * Wl[(long)c * CC + d];
                outb[d] = sinf(30.f * a);
            }
            __syncthreads();
            const float* ni = outb;
            outb = (outb == bufA) ? bufB : bufA;
            inb = ni;
        }
        for (int half = 0; half < 2; ++half) {
            int d = tid + half * 256;
            float a = nBov[d];
            for (int c = 0; c < CC; ++c) a += inb[c] * nWov[(long)c * CC + d];
            outb[d] = a;
        }
        __syncthreads();
        // --- unit norm -> cur (pre-gate hh) ---
        red[tid] = outb[tid] * outb[tid] + outb[tid + 256] * outb[tid + 256];
        __syncthreads();
        for (int off = 128; off > 0; off >>= 1) {
            if (tid < off) red[tid] += red[tid + off];
            __syncthreads();
        }
        float invn = 1.f / (sqrtf(red[0]) + 1e-8f);
        __syncthreads();
        cur[tid] = outb[tid] * invn; cur[tid + 256] = outb[tid + 256] * invn;
        __syncthreads();
        // --- g-MLP: 3 hidden (lrelu) + Cx2 out ---
        const float* gi = cur; float* go = bufA;
        for (int layer = 0; layer < LL; ++layer) {
            const float* Wl = gWv + (long)layer * CC * CC;
            const float* Bl = gBv + layer * CC;
            for (int half = 0; half < 2; ++half) {
                int d = tid + half * 256;
                float a = Bl[d];
                for (int c = 0; c < CC; ++c) a += gi[c] * Wl[(long)c * CC + d];
                go[d] = (a >= 0.f) ? a : 0.2f * a;
            }
            __syncthreads();
            const float* ni = go;
            go = (go == bufA) ? bufB : bufA;
            gi = ni;
        }
        float p0 = 0.f, p1 = 0.f;
        for (int c = tid; c < CC; c += 256) {
            float hv = gi[c];
            p0 += hv * gWov[c * 2 + 0];
            p1 += hv * gWov[c * 2 + 1];
        }
        red[tid] = p0; __syncthreads();
        for (int off = 128; off > 0; off >>= 1) {
            if (tid < off) red[tid] += red[tid + off];
            __syncthreads();
        }
        if (tid == 0) sc[0] = red[0] + gBov[0];
        __syncthreads();
        red[tid] = p1; __syncthreads();
        for (int off = 128; off > 0; off >>= 1) {
            if (tid < off) red[tid] += red[tid + off];
            __syncthreads();
        }
        if (tid == 0) sc[1] = red[0] + gBov[1];
        __syncthreads();
        float l0 = sc[0], l1 = sc[1];
        float g0 = 1.f / (1.f + expf(l1 - l0));      // softmax prob of gate 0
        cur[tid] *= g0; cur[tid + 256] *= g0;        // hh = hh * g[:, :1]
        long base = ((long)t * KK + j) * CC;
        res[base + tid] = cur[tid];
        res[base + tid + 256] = cur[tid + 256];
        if (tid == 0) {
            validOut[t * KK + j] = (aliveS && (j < maxit)) ? 1.f : 0.f;
            aliveS = aliveS && (l0 >= l1);           // argmax(g) != 1
        }
        __syncthreads();
    }
}

// ---------------------------------------------------------------------------
// Twiddle tables: CT/ST [512,257] (n-major, B for forward DFT),
// COSB/SINB [257,512] (f-major, B for inverse DFT). Angle reduced exactly.
// ---------------------------------------------------------------------------
__global__ void twiddle_kernel(float* CT, float* ST, float* COSB, float* SINB)
{
    int idx = blockIdx.x * blockDim.x + threadIdx.x;
    if (idx < 512 * NC) {
        int n = idx / NC, f = idx % NC;
        int m = (n * f) & 511;
        float ang = (float)m * (6.283185307179586f / 512.f);
        float c = cosf(ang), s = sinf(ang);
        CT[n * NC + f] = c;  ST[n * NC + f] = s;
        COSB[f * 512 + n] = c;  SINB[f * 512 + n] = s;
    }
}

// G = (w_f/N)*coeffs*valid*R ; H = (w_f/N)*coeffs*valid*Ineg   (in place)
__global__ void ghmul_kernel(const float* __restrict__ coeffs,
                             const float* __restrict__ valid,
                             float* __restrict__ R, float* __restrict__ I)
{
    int idx = blockIdx.x * blockDim.x + threadIdx.x;
    if (idx < 2048 * NC) {
        int row = idx / NC, f = idx % NC;
        float w = (f == 0 || f == 256) ? (1.f / 512.f) : (2.f / 512.f);
        float sc = w * coeffs[idx] * valid[row];
        R[idx] *= sc;
        I[idx] *= sc;
    }
}

// 50%-overlap add of the 2*SPF noise frames -> noise[T, 4096]
__global__ void ola_kernel(const float* __restrict__ nf, float* __restrict__ out)
{
    long idx = (long)blockIdx.x * blockDim.x + threadIdx.x;
    if (idx < (long)TT * FRLEN) {
        int t = (int)(idx / FRLEN);
        int rem = (int)(idx % FRLEN);
        int j = rem >> 8, s = rem & 255;
        float a = nf[((long)t * KK + j) * 512 + s];
        float b = (j > 0) ? nf[((long)t * KK + j - 1) * 512 + 256 + s] : 0.f;
        out[idx] = a + b;
    }
}

// ---------------------------------------------------------------------------
// Oscillator bank + silence mask + route gain -> sig[T, 4096]
// One workgroup per frame; block-wide inclusive scan for the phase cumsum.
// ---------------------------------------------------------------------------
__global__ __launch_bounds__(256)
void osc_sig_kernel(const float* __restrict__ op, const float* __restrict__ valid,
                    const float* __restrict__ noise, const int* __restrict__ sil,
                    const float* __restrict__ rr, float* __restrict__ sig)
{
    __shared__ float ampk[KK], frk[KK];
    __shared__ float part[256];
    int t = blockIdx.x, tid = threadIdx.x;
    float acc[16];
#pragma unroll
    for (int e = 0; e < 16; ++e) acc[e] = 0.f;

    for (int o = 0; o < NOSC; ++o) {
        if (tid < KK) {
            long b = ((long)t * KK + tid) * 128 + 2 * o;
            float xx = op[b], yy = op[b + 1];
            ampk[tid] = sqrtf(xx * xx + yy * yy + 1e-12f) * valid[t * KK + tid];
            frk[tid]  = atan2f(yy, xx);          // = freq*pi directly
        }
        __syncthreads();
        float pv[16];
        float running = 0.f;
#pragma unroll
        for (int e = 0; e < 16; ++e) {
            int i = tid * 16 + e;
            float pos = (i + 0.5f) * (1.f / 256.f) - 0.5f;
            float fl = floorf(pos);
            float w = pos - fl;
            int i0 = clampi((int)fl, 15), i1 = clampi((int)fl + 1, 15);
            running += frk[i0] * (1.f - w) + frk[i1] * w;
            pv[e] = running;
        }
        part[tid] = running;
        __syncthreads();
        for (int off = 1; off < 256; off <<= 1) {
            float add = (tid >= off) ? part[tid - off] : 0.f;
            __syncthreads();
            part[tid] += add;
            __syncthreads();
        }
        float base = (tid > 0) ? part[tid - 1] : 0.f;
#pragma unroll
        for (int e = 0; e < 16; ++e) {
            int i = tid * 16 + e;
            float pos = (i + 0.5f) * (1.f / 256.f) - 0.5f;
            float fl = floorf(pos);
            float w = pos - fl;
            int i0 = clampi((int)fl, 15), i1 = clampi((int)fl + 1, 15);
            float a = ampk[i0] * (1.f - w) + ampk[i1] * w;
            acc[e] += sinf(base + pv[e]) * a;
        }
        __syncthreads();
    }
    int issil = sil[t];
    float rv = rr[t];
    int maxit = TT - t;
#pragma unroll
    for (int e = 0; e < 16; ++e) {
        int i = tid * 16 + e;
        int jj = i >> 8;
        float s;
        if (issil) s = (jj < maxit) ? 1e-8f : 0.f;
        else       s = acc[e] + noise[(long)t * FRLEN + i];
        sig[(long)t * FRLEN + i] = s * rv;
    }
}

// segment-sum (gather form): dry[p] = sum over overlapping frames
__global__ void dry_kernel(const float* __restrict__ sig, float* __restrict__ dry)
{
    int p = blockIdx.x * blockDim.x + threadIdx.x;
    if (p < TOTAL) {
        float s = 0.f;
        int tb = p >> 8, pm = p & 255;
#pragma unroll
        for (int dt = 0; dt < KK; ++dt) {
            int t = tb - dt;
            if (t >= 0 && t < TT) s += sig[(long)t * FRLEN + pm + (dt << 8)];
        }
        dry[p] = s;
    }
}

__global__ void imp_kernel(const float* __restrict__ rooms,
                           const float* __restrict__ ir, float* __restrict__ imp)
{
    int p = blockIdx.x * blockDim.x + threadIdx.x;
    if (p < TOTAL) {
        float s = 0.f;
#pragma unroll
        for (int v = 0; v < 8; ++v) s += rooms[v] * ir[(long)v * TOTAL + p];
        imp[p] = s;
    }
}

// FFT(65536) conv of two 32768 signals, first 32768 samples == causal
// triangular linear convolution.
__global__ void conv_kernel(const float* __restrict__ dry,
                            const float* __restrict__ imp, float* __restrict__ wet)
{
    int p = blockIdx.x * blockDim.x + threadIdx.x;
    if (p < TOTAL) {
        float s = 0.f;
        for (int q = 0; q <= p; ++q) s += dry[q] * imp[p - q];
        wet[p] = s;
    }
}

__global__ void final_kernel(const float* __restrict__ dry,
                             const float* __restrict__ wet,
                             const float* __restrict__ mixv, float* __restrict__ out)
{
    int p = blockIdx.x * blockDim.x + threadIdx.x;
    if (p < TOTAL) {
        float m = mixv[0];
        out[p] = dry[p] * m + wet[p] * (1.f - m);
    }
}

// ---------------------------------------------------------------------------
// Host-side orchestration
// ---------------------------------------------------------------------------
static void gemm(hipStream_t s, const float* A, int lda, long aB,
                 const float* W, int ldb, long wB, const int* vsel,
                 const float* bias, long biasB,
                 float* Out, int ldo, long oB,
                 int M, int N, int K, int act, int accMode, int batch)
{
    dim3 g((N + 15) / 16, (M + 15) / 16, batch);
    wmma_gemm_f32<<<g, 32, 0, s>>>(A, lda, aB, W, ldb, wB, vsel,
                                   bias, biasB, Out, ldo, oB,
                                   M, N, K, act, accMode);
}

extern "C" void kernel_launch(void* const* d_in, const int* in_sizes, int n_in,
                              void* d_out, int out_size, void* d_ws, size_t ws_size,
                              hipStream_t stream)
{
    (void)in_sizes; (void)n_in; (void)out_size; (void)ws_size;
    const float* x    = (const float*)d_in[0];
    const float* wn   = (const float*)d_in[1];
    const float* nW   = (const float*)d_in[2];
    const float* nB   = (const float*)d_in[3];
    const float* nWo  = (const float*)d_in[4];
    const float* nBo  = (const float*)d_in[5];
    const float* gW   = (const float*)d_in[6];
    const float* gB   = (const float*)d_in[7];
    const float* gWo  = (const float*)d_in[8];
    const float* gBo  = (const float*)d_in[9];
    const float* zW   = (const float*)d_in[10];
    const float* zB   = (const float*)d_in[11];
    const float* zWo  = (const float*)d_in[12];
    const float* zBo  = (const float*)d_in[13];
    const float* oW   = (const float*)d_in[14];
    const float* oB   = (const float*)d_in[15];
    const float* oWo  = (const float*)d_in[16];
    const float* oBo  = (const float*)d_in[17];
    const float* cW   = (const float*)d_in[18];
    const float* cB   = (const float*)d_in[19];
    const float* cWo  = (const float*)d_in[20];
    const float* cBo  = (const float*)d_in[21];
    const float* rW   = (const float*)d_in[22];
    const float* rB   = (const float*)d_in[23];
    const float* rWo  = (const float*)d_in[24];
    const float* rBo  = (const float*)d_in[25];
    const float* rmW  = (const float*)d_in[26];
    const float* rmB  = (const float*)d_in[27];
    const float* rmWo = (const float*)d_in[28];
    const float* rmBo = (const float*)d_in[29];
    const float* mxW  = (const float*)d_in[30];
    const float* mxB  = (const float*)d_in[31];
    const float* mxWo = (const float*)d_in[32];
    const float* mxBo = (const float*)d_in[33];
    const float* vir  = (const float*)d_in[34];
    float* out = (float*)d_out;

    // ---- workspace carving (floats) ----
    float* ws = (float*)d_ws;
    size_t off = 0;
    auto alloc = [&](size_t n) { float* p = ws + off; off += n; return p; };
    float* z      = alloc(CC);
    float* s1     = alloc(CC);
    float* s2     = alloc(CC);
    float* rmlog  = alloc(16);
    float* mxlog  = alloc(16);
    float* rooms  = alloc(8);
    float* mixv   = alloc(8);
    float* bufA   = alloc((size_t)TT * CC);
    float* bufB   = alloc((size_t)TT * CC);
    float* hbuf   = alloc((size_t)TT * CC);
    float* rlog   = alloc((size_t)TT * 16);
    int*   vv     = (int*)alloc(TT);
    int*   sil    = (int*)alloc(TT);
    float* rr     = alloc(TT);
    float* res    = alloc((size_t)TT * KK * CC);
    float* valid  = alloc((size_t)TT * KK);
    float* zb1    = alloc((size_t)TT * KK * CC);
    float* zb2    = alloc((size_t)TT * KK * CC);
    float* coeffs = alloc((size_t)TT * KK * NC);
    float* opb    = alloc((size_t)TT * KK * 128);
    float* CTt    = alloc((size_t)512 * NC);
    float* STt    = alloc((size_t)512 * NC);
    float* COSB   = alloc((size_t)NC * 512);
    float* SINB   = alloc((size_t)NC * 512);
    float* Rb     = alloc((size_t)TT * KK * NC);
    float* Ib     = alloc((size_t)TT * KK * NC);
    float* nf     = alloc((size_t)TT * KK * 512);
    float* noiseb = alloc((size_t)TT * FRLEN);
    float* sigb   = alloc((size_t)TT * FRLEN);
    float* dry    = alloc(TOTAL);
    float* imp    = alloc(TOTAL);
    float* wet    = alloc(TOTAL);

    const long CxC = (long)CC * CC;

    // twiddles (independent, launch first)
    twiddle_kernel<<<(512 * NC + 255) / 256, 256, 0, stream>>>(CTt, STt, COSB, SINB);

    // ---- global heads: z, rooms, mix ----
    zmean_kernel<<<2, 256, 0, stream>>>(x, z);
    gemm(stream, z, CC, 0, rmW,            CC, 0, nullptr, rmB,           0, s1,    CC, 0, 1, CC, CC, ACT_LRELU, 0, 1);
    gemm(stream, s1, CC, 0, rmW + CxC,     CC, 0, nullptr, rmB + CC,      0, s2,    CC, 0, 1, CC, CC, ACT_LRELU, 0, 1);
    gemm(stream, s2, CC, 0, rmW + 2 * CxC, CC, 0, nullptr, rmB + 2 * CC,  0, s1,    CC, 0, 1, CC, CC, ACT_LRELU, 0, 1);
    gemm(stream, s1, CC, 0, rmWo,           8, 0, nullptr, rmBo,          0, rmlog,  8, 0, 1,  8, CC, ACT_NONE,  0, 1);
    gemm(stream, z, CC, 0, mxW,            CC, 0, nullptr, mxB,           0, s1,    CC, 0, 1, CC, CC, ACT_LRELU, 0, 1);
    gemm(stream, s1, CC, 0, mxW + CxC,     CC, 0, nullptr, mxB + CC,      0, s2,    CC, 0, 1, CC, CC, ACT_LRELU, 0, 1);
    gemm(stream, s2, CC, 0, mxW + 2 * CxC, CC, 0, nullptr, mxB + 2 * CC,  0, s1,    CC, 0, 1, CC, CC, ACT_LRELU, 0, 1);
    gemm(stream, s1, CC, 0, mxWo,           1, 0, nullptr, mxBo,          0, mxlog,  1, 0, 1,  1, CC, ACT_NONE,  0, 1);
    heads_kernel<<<1, 32, 0, stream>>>(rmlog, mxlog, rooms, mixv);

    // ---- conductor MLP: h = cMLP(x[0])  (128x512 GEMMs, WMMA) ----
    gemm(stream, x,    CC, 0, cW,            CC, 0, nullptr, cB,          0, bufA, CC, 0, TT, CC, CC, ACT_LRELU, 0, 1);
    gemm(stream, bufA, CC, 0, cW + CxC,      CC, 0, nullptr, cB + CC,     0, bufB, CC, 0, TT, CC, CC, ACT_LRELU, 0, 1);
    gemm(stream, bufB, CC, 0, cW + 2 * CxC,  CC, 0, nullptr, cB + 2 * CC, 0, bufA, CC, 0, TT, CC, CC, ACT_LRELU, 0, 1);
    gemm(stream, bufA, CC, 0, cWo,           CC, 0, nullptr, cBo,         0, hbuf, CC, 0, TT, CC, CC, ACT_NONE,  0, 1);

    // ---- routing ----
    gemm(stream, hbuf, CC, 0, rW,            CC, 0, nullptr, rB,          0, bufA, CC, 0, TT, CC, CC, ACT_LRELU, 0, 1);
    gemm(stream, bufA, CC, 0, rW + CxC,      CC, 0, nullptr, rB + CC,     0, bufB, CC, 0, TT, CC, CC, ACT_LRELU, 0, 1);
    gemm(stream, bufB, CC, 0, rW + 2 * CxC,  CC, 0, nullptr, rB + 2 * CC, 0, bufA, CC, 0, TT, CC, CC, ACT_LRELU, 0, 1);
    gemm(stream, bufA, CC, 0, rWo,            9, 0, nullptr, rBo,         0, rlog,  9, 0, TT,  9, CC, ACT_NONE,  0, 1);
    route_kernel<<<1, 128, 0, stream>>>(rlog, vv, sil, rr);

    // ---- K=16 recurrence (VALU matvecs, voice-gathered weights) ----
    scan_kernel<<<TT, 256, 0, stream>>>(hbuf, vv, nW, nB, nWo, nBo,
                                        gW, gB, gWo, gBo, res, valid);

    // ---- per-frame z-MLP: 128 batched 16x512 @ 512x512 WMMA GEMMs ----
    const long LWB = (long)LL * CxC, LBB = (long)LL * CC;
    gemm(stream, res, CC, (long)KK * CC, zW,            CC, LWB, vv, zB,          LBB, zb1,    CC, (long)KK * CC, KK, CC, CC, ACT_LRELU, 0, TT);
    gemm(stream, zb1, CC, (long)KK * CC, zW + CxC,      CC, LWB, vv, zB + CC,     LBB, zb2,    CC, (long)KK * CC, KK, CC, CC, ACT_LRELU, 0, TT);
    gemm(stream, zb2, CC, (long)KK * CC, zW + 2 * CxC,  CC, LWB, vv, zB + 2 * CC, LBB, zb1,    CC, (long)KK * CC, KK, CC, CC, ACT_LRELU, 0, TT);
    gemm(stream, zb1, CC, (long)KK * CC, zWo,           NC, (long)CC * NC, vv, zBo, NC, coeffs, NC, (long)KK * NC, KK, NC, CC, ACT_NONE,  0, TT);

    // ---- per-frame o-MLP ----
    gemm(stream, res, CC, (long)KK * CC, oW,            CC, LWB, vv, oB,          LBB, zb1, CC, (long)KK * CC, KK, CC,  CC, ACT_LRELU, 0, TT);
    gemm(stream, zb1, CC, (long)KK * CC, oW + CxC,      CC, LWB, vv, oB + CC,     LBB, zb2, CC, (long)KK * CC, KK, CC,  CC, ACT_LRELU, 0, TT);
    gemm(stream, zb2, CC, (long)KK * CC, oW + 2 * CxC,  CC, LWB, vv, oB + 2 * CC, LBB, zb1, CC, (long)KK * CC, KK, CC,  CC, ACT_LRELU, 0, TT);
    gemm(stream, zb1, CC, (long)KK * CC, oWo,          128, (long)CC * 128, vv, oBo, 128, opb, 128, (long)KK * 128, KK, 128, CC, ACT_NONE, 0, TT);

    // ---- noise branch: rfft (GEMM vs twiddles), spectral scale, irfft ----
    gemm(stream, wn, 512, 0, CTt, NC, 0, nullptr, nullptr, 0, Rb, NC, 0, TT * KK, NC, 512, ACT_NONE, 0, 1);
    gemm(stream, wn, 512, 0, STt, NC, 0, nullptr, nullptr, 0, Ib, NC, 0, TT * KK, NC, 512, ACT_NONE, 0, 1);
    ghmul_kernel<<<(TT * KK * NC + 255) / 256, 256, 0, stream>>>(coeffs, valid, Rb, Ib);
    gemm(stream, Rb, NC, 0, COSB, 512, 0, nullptr, nullptr, 0, nf, 512, 0, TT * KK, 512, NC, ACT_NONE, 0, 1);
    gemm(stream, Ib, NC, 0, SINB, 512, 0, nullptr, nullptr, 0, nf, 512, 0, TT * KK, 512, NC, ACT_NONE, 1, 1);
    ola_kernel<<<(TT * FRLEN + 255) / 256, 256, 0, stream>>>(nf, noiseb);

    // ---- oscillator bank + mask + gain ----
    osc_sig_kernel<<<TT, 256, 0, stream>>>(opb, valid, noiseb, sil, rr, sigb);

    // ---- overlap-add to dry, reverb, mix ----
    dry_kernel<<<(TOTAL + 255) / 256, 256, 0, stream>>>(sigb, dry);
    imp_kernel<<<(TOTAL + 255) / 256, 256, 0, stream>>>(rooms, vir, imp);
    conv_kernel<<<(TOTAL + 255) / 256, 256, 0, stream>>>(dry, imp, wet);
    final_kernel<<<(TOTAL + 255) / 256, 256, 0, stream>>>(dry, wet, mixv, out);
}